// GINE_8426725834833
// MI455X (gfx1250) — compile-verified
//
#include <hip/hip_runtime.h>
#include <math.h>

typedef __attribute__((ext_vector_type(2))) float v2f;
typedef __attribute__((ext_vector_type(8))) float v8f;

#define NEGS 0.01f
#define BN_SCALE 0.99999500003749971f   // 1/sqrt(1+1e-5)
#define KCHUNK 256
#define LDA_PAD 260                     // 260 % 64 == 4 -> conflict-free row skew

__global__ __launch_bounds__(256) void k_zero(float* __restrict__ p, int n) {
    int i = blockIdx.x * blockDim.x + threadIdx.x;
    if (i < n) p[i] = 0.f;
}

// Edge message + scatter-add:  aggr[dst] += relu(h[src] + edge_attr @ We + be)
__global__ __launch_bounds__(256) void k_edge(
    const float* __restrict__ h, int ldh, int cin,
    const float* __restrict__ edge_attr,
    const float* __restrict__ We, const float* __restrict__ be,
    const int* __restrict__ src, const int* __restrict__ dst,
    float* __restrict__ aggr, int E)
{
    __shared__ float sWe[16 * 256];
    __shared__ float sbe[256];
    for (int i = threadIdx.x; i < 16 * cin; i += 256) sWe[i] = We[i];
    for (int i = threadIdx.x; i < cin; i += 256) sbe[i] = be[i];
    __syncthreads();

    const int wave = threadIdx.x >> 5;
    const int lane = threadIdx.x & 31;
    const int EPW  = 4;
    int e0 = (blockIdx.x * 8 + wave) * EPW;

    for (int ei = 0; ei < EPW; ++ei) {
        int e = e0 + ei;
        if (e >= E) break;
        float ea[16];
#pragma unroll
        for (int k = 0; k < 16; ++k) ea[k] = edge_attr[(long)e * 16 + k];
        int s = src[e], d = dst[e];
        const float* hrow = h + (long)s * ldh;
        for (int j = lane; j < cin; j += 32) {
            float acc = sbe[j];
#pragma unroll
            for (int k = 0; k < 16; ++k) acc = fmaf(ea[k], sWe[k * cin + j], acc);
            float m = hrow[j] + acc;
            m = m > 0.f ? m : 0.f;
            atomicAdd(&aggr[(long)d * cin + j], m);
        }
    }
}

// Fused GEMM: out = epilogue( ((1+eps)*A + aggr) @ W + b )   [mode 0: GIN+BN+leaky^2]
//             out = leaky( A @ W + b )                       [mode 1: classifier]
// Block = 8 waves; each wave computes a 16x32 f32 tile (2 accumulators) via
// V_WMMA_F32_16X16X4_F32, K-loop unrolled x4 so 8 WMMAs issue per body.
__global__ __launch_bounds__(256) void k_gemm(
    const float* __restrict__ A, int lda,
    const float* __restrict__ aggr,     // nullptr in mode 1; ld == K
    const float* __restrict__ epsp,     // nullptr in mode 1
    const float* __restrict__ W,        // K x cout, row-major
    const float* __restrict__ bias,
    const float* __restrict__ gamma, const float* __restrict__ beta,
    float* __restrict__ out, int ldo,
    int K, int cout, int mode)
{
    __shared__ float sA[16 * LDA_PAD];
    const int lane = threadIdx.x & 31;
    const int wave = threadIdx.x >> 5;
    const long rowBase = (long)blockIdx.x * 16;
    const int  colBase = (blockIdx.y * 8 + wave) * 32;
    const bool active  = (colBase < cout);   // wave-uniform -> EXEC all-1s inside

    float epsv = 1.f;
    if (epsp) epsv = 1.0f + *epsp;

    v8f acc0 = {}, acc1 = {};
    for (int kb = 0; kb < K; kb += KCHUNK) {
        int chunk = K - kb; if (chunk > KCHUNK) chunk = KCHUNK;
        // stage combined A panel (16 x chunk) into LDS; shared by all 8 waves
        for (int i = threadIdx.x; i < 16 * chunk; i += 256) {
            int r = i / chunk, c = i % chunk;
            long row = rowBase + r;
            float v = A[row * (long)lda + kb + c];
            if (aggr) v = epsv * v + aggr[row * (long)K + kb + c];
            sA[r * LDA_PAD + c] = v;
        }
        __syncthreads();

        if (active) {
            const int koff = (lane >= 16) ? 2 : 0;   // ISA 16x4 f32 A layout
            const int arow = lane & 15;
            const int col0 = colBase + (lane & 15);
#pragma unroll 4
            for (int kk = 0; kk < chunk; kk += 4) {
                v2f a, b0, b1;
                a.x = sA[arow * LDA_PAD + kk + koff];
                a.y = sA[arow * LDA_PAD + kk + koff + 1];
                long wb = (long)(kb + kk + koff) * cout + col0;
                b0.x = W[wb];
                b0.y = W[wb + cout];
                b1.x = W[wb + 16];
                b1.y = W[wb + cout + 16];
                acc0 = __builtin_amdgcn_wmma_f32_16x16x4_f32(
                    false, a, false, b0, (short)0, acc0, false, false);
                acc1 = __builtin_amdgcn_wmma_f32_16x16x4_f32(
                    false, a, false, b1, (short)0, acc1, false, false);
            }
        }
        __syncthreads();
    }

    if (active) {
        const int rOff = (lane >= 16) ? 8 : 0;       // C/D VGPR->row mapping
#pragma unroll
        for (int t = 0; t < 2; ++t) {
            const int col = colBase + (lane & 15) + t * 16;
            float bv = bias[col];
            float gv = 0.f, btv = 0.f;
            if (mode == 0) { gv = gamma[col]; btv = beta[col]; }
#pragma unroll
            for (int r = 0; r < 8; ++r) {
                float v = (t == 0 ? acc0[r] : acc1[r]) + bv;
                if (mode == 0) {
                    v = gv * (v * BN_SCALE) + btv;
                    v = v >= 0.f ? v : NEGS * v;   // leaky inside Sequential
                    v = v >= 0.f ? v : NEGS * v;   // outer leaky in forward
                } else {
                    v = v >= 0.f ? v : NEGS * v;
                }
                long row = rowBase + r + rOff;
                out[row * (long)ldo + col] = v;
            }
        }
    }
}

__global__ __launch_bounds__(256) void k_pool_scatter(
    const float* __restrict__ hcat, const int* __restrict__ batch,
    float* __restrict__ pool)
{
    int n = blockIdx.x, t = threadIdx.x;
    float v = hcat[(long)n * 1024 + 512 + t];      // h3 slot
    atomicAdd(&pool[(long)batch[n] * 256 + t], v);
}

__global__ __launch_bounds__(256) void k_pool_gather(
    float* __restrict__ hcat, const int* __restrict__ batch,
    const float* __restrict__ pool)
{
    int n = blockIdx.x, t = threadIdx.x;
    hcat[(long)n * 1024 + 768 + t] = pool[(long)batch[n] * 256 + t];
}

__global__ __launch_bounds__(256) void k_final(
    const float* __restrict__ A,      // N x 128
    const float* __restrict__ Wf, const float* __restrict__ bf,
    float* __restrict__ out, int n)
{
    int i = blockIdx.x * blockDim.x + threadIdx.x;
    if (i >= n) return;
    const float* row = A + (long)i * 128;
    float acc = bf[0];
#pragma unroll 8
    for (int k = 0; k < 128; ++k) acc = fmaf(row[k], Wf[k], acc);
    out[i] = 1.0f / (1.0f + expf(-acc));
}

extern "C" void kernel_launch(void* const* d_in, const int* in_sizes, int n_in,
                              void* d_out, int out_size, void* d_ws, size_t ws_size,
                              hipStream_t stream) {
    const int N = 100000, E = 400000, M = 5000;
    const float* x         = (const float*)d_in[0];
    const int*   eidx      = (const int*)d_in[1];
    const float* edge_attr = (const float*)d_in[2];
    const int*   batch     = (const int*)d_in[3];
    const int*   src = eidx;        // edge_index[0]
    const int*   dst = eidx + E;    // edge_index[1]

    const float *cWe[3], *cbe[3], *cW[3], *cb[3], *cg[3], *cbt[3], *ceps[3];
    for (int l = 0; l < 3; ++l) {
        int b = 4 + 7 * l;          // {We, be, W, b, gamma, beta, eps}
        cWe[l]  = (const float*)d_in[b];
        cbe[l]  = (const float*)d_in[b + 1];
        cW[l]   = (const float*)d_in[b + 2];
        cb[l]   = (const float*)d_in[b + 3];
        cg[l]   = (const float*)d_in[b + 4];
        cbt[l]  = (const float*)d_in[b + 5];
        ceps[l] = (const float*)d_in[b + 6];
    }
    const float* W0c = (const float*)d_in[25];
    const float* b0c = (const float*)d_in[26];
    const float* W1c = (const float*)d_in[27];
    const float* b1c = (const float*)d_in[28];
    const float* Wf  = (const float*)d_in[29];
    const float* bf  = (const float*)d_in[30];

    float* ws   = (float*)d_ws;
    float* aggr = ws;                         // N*256
    float* hcat = aggr + (size_t)N * 256;     // N*1024: [h1|h2|h3|pooled]
    float* pool = hcat + (size_t)N * 1024;    // M*256
    float* c1   = pool + (size_t)M * 256;     // N*256
    float* c2   = c1   + (size_t)N * 256;     // N*128

    dim3 blk(256);

    // conv layer 0: c_in=64 -> 256
    k_zero<<<dim3((N * 64 + 255) / 256), blk, 0, stream>>>(aggr, N * 64);
    k_edge<<<dim3(E / 32), blk, 0, stream>>>(x, 64, 64, edge_attr, cWe[0], cbe[0],
                                             src, dst, aggr, E);
    k_gemm<<<dim3(N / 16, 1), blk, 0, stream>>>(x, 64, aggr, ceps[0], cW[0], cb[0],
                                                cg[0], cbt[0], hcat + 0, 1024, 64, 256, 0);
    // conv layer 1: 256 -> 256
    k_zero<<<dim3((N * 256 + 255) / 256), blk, 0, stream>>>(aggr, N * 256);
    k_edge<<<dim3(E / 32), blk, 0, stream>>>(hcat + 0, 1024, 256, edge_attr, cWe[1], cbe[1],
                                             src, dst, aggr, E);
    k_gemm<<<dim3(N / 16, 1), blk, 0, stream>>>(hcat + 0, 1024, aggr, ceps[1], cW[1], cb[1],
                                                cg[1], cbt[1], hcat + 256, 1024, 256, 256, 0);
    // conv layer 2: 256 -> 256
    k_zero<<<dim3((N * 256 + 255) / 256), blk, 0, stream>>>(aggr, N * 256);
    k_edge<<<dim3(E / 32), blk, 0, stream>>>(hcat + 256, 1024, 256, edge_attr, cWe[2], cbe[2],
                                             src, dst, aggr, E);
    k_gemm<<<dim3(N / 16, 1), blk, 0, stream>>>(hcat + 256, 1024, aggr, ceps[2], cW[2], cb[2],
                                                cg[2], cbt[2], hcat + 512, 1024, 256, 256, 0);
    // global add-pool + broadcast back
    k_zero<<<dim3((M * 256 + 255) / 256), blk, 0, stream>>>(pool, M * 256);
    k_pool_scatter<<<dim3(N), blk, 0, stream>>>(hcat, batch, pool);
    k_pool_gather<<<dim3(N), blk, 0, stream>>>(hcat, batch, pool);
    // classifier
    k_gemm<<<dim3(N / 16, 1), blk, 0, stream>>>(hcat, 1024, nullptr, nullptr, W0c, b0c,
                                                nullptr, nullptr, c1, 256, 1024, 256, 1);
    k_gemm<<<dim3(N / 16, 1), blk, 0, stream>>>(c1, 256, nullptr, nullptr, W1c, b1c,
                                                nullptr, nullptr, c2, 128, 256, 128, 1);
    k_final<<<dim3((N + 255) / 256), blk, 0, stream>>>(c2, Wf, bf, (float*)d_out, N);
}